// YOLOV3Decoder_19645180412545
// MI455X (gfx1250) — compile-verified
//
#include <hip/hip_runtime.h>
#include <stdint.h>

#define IMG_MAX  607
#define TOP_N    1000
#define MIN_SCOREF 0.05f
#define NMS_THRF  0.5f
#define MAX_DET  100
#define BATCH    8
#define NCLS     80
#define N0 1083
#define N1 4332
#define N2 17328
#define NTOT (N0 + N1 + N2)   /* 22743 */
#define P0 2048
#define P1 8192
#define P2 32768
#define PG 4096
#define NSEL (3 * TOP_N)      /* 3000 */
#define SORT_TILE 4096

#if defined(__HIP_DEVICE_COMPILE__) && defined(__gfx1250__) && \
    __has_builtin(__builtin_amdgcn_global_load_async_to_lds_b32)
#define USE_ASYNC_LDS 1
#else
#define USE_ASYNC_LDS 0
#endif

typedef __attribute__((address_space(1))) int* as1_i32p;
typedef __attribute__((address_space(3))) int* as3_i32p;

__device__ __forceinline__ float sigf(float x) {
    return 1.0f / (1.0f + expf(-x));
}

// Descending-sort key: higher score => larger key; equal score => lower index larger key.
__device__ __forceinline__ unsigned long long make_key(float s, unsigned idx) {
    unsigned u = __float_as_uint(s);
    u = (u & 0x80000000u) ? ~u : (u | 0x80000000u);
    return ((unsigned long long)u << 32) | (unsigned)(0xFFFFFFFFu - idx);
}

// ---------------- decode one pyramid level ----------------
__global__ void decode_kernel(const float* __restrict__ obj,
                              const float* __restrict__ reg,
                              const float* __restrict__ cls,
                              const float* __restrict__ anc,
                              float* __restrict__ sc, int* __restrict__ cl,
                              float* __restrict__ bx,
                              int N, int lvlOff) {
    int t = blockIdx.x * blockDim.x + threadIdx.x;
    if (t >= BATCH * N) return;
    int b = t / N, i = t - b * N;

    const float4* cr4 = (const float4*)(cls + (size_t)t * NCLS);
    __builtin_prefetch(cr4 + 4, 0, 0);   // gfx1250 global_prefetch_b8
    float best = -3.0e38f; int bi = 0;
    #pragma unroll
    for (int q = 0; q < NCLS / 4; ++q) {
        float4 v = cr4[q]; int j = q * 4;
        if (v.x > best) { best = v.x; bi = j; }
        if (v.y > best) { best = v.y; bi = j + 1; }
        if (v.z > best) { best = v.z; bi = j + 2; }
        if (v.w > best) { best = v.w; bi = j + 3; }
    }
    float score = sigf(best) * sigf(obj[t]);

    float4 r = ((const float4*)reg)[t];
    const float* a = anc + (size_t)t * 5;
    float ax = a[0], ay = a[1], aw = a[2], ah = a[3], as_ = a[4];
    float cx = (sigf(r.x) + ax) * as_;
    float cy = (sigf(r.y) + ay) * as_;
    float w  = expf(r.z) * aw / as_;
    float h  = expf(r.w) * ah / as_;
    float x1f = cx - 0.5f * w, y1f = cy - 0.5f * h;
    float x2f = w + x1f,       y2f = h + y1f;
    int x1 = (int)x1f, y1 = (int)y1f, x2 = (int)x2f, y2 = (int)y2f; // trunc toward 0
    x1 = max(x1, 0); y1 = max(y1, 0);
    x2 = min(x2, IMG_MAX); y2 = min(y2, IMG_MAX);

    size_t o = (size_t)b * NTOT + lvlOff + i;
    sc[o] = score; cl[o] = bi;
    float4 bb; bb.x = (float)x1; bb.y = (float)y1; bb.z = (float)x2; bb.w = (float)y2;
    ((float4*)bx)[o] = bb;
}

// ---------------- sort-key init (per-level, padded) ----------------
__global__ void key_init_kernel(const float* __restrict__ sc,
                                unsigned long long* __restrict__ keys,
                                int P, int N, int lvlOff) {
    int t = blockIdx.x * blockDim.x + threadIdx.x;
    if (t >= BATCH * P) return;
    int b = t / P, p = t - b * P;
    unsigned long long k = 0ull;
    if (p < N) k = make_key(sc[(size_t)b * NTOT + lvlOff + p], (unsigned)p);
    keys[t] = k;
}

// ---------------- LDS-fused bitonic: full local sort (phases k=2..tile) ----------------
// One block sorts one `tile`-element span entirely in LDS; direction bits use the
// segment-local index so this is bit-identical to the monolithic bitonic network.
__global__ __launch_bounds__(1024) void lds_sort_kernel(unsigned long long* __restrict__ keys,
                                                        int P, int tile) {
    __shared__ unsigned long long sk[SORT_TILE];
    const int tid = threadIdx.x;
    const size_t base = (size_t)blockIdx.x * tile;
    const int segBase = (int)(base & (size_t)(P - 1));
    for (int e = tid; e < tile; e += 1024) sk[e] = keys[base + e];
    __syncthreads();
    for (int k = 2; k <= tile; k <<= 1) {
        for (int j = k >> 1; j >= 1; j >>= 1) {
            for (int p = tid; p < (tile >> 1); p += 1024) {
                int i = ((p & ~(j - 1)) << 1) | (p & (j - 1));
                int ixj = i | j;
                unsigned long long a = sk[i], c = sk[ixj];
                bool up = (((segBase + i) & k) == 0);   // descending region
                bool sw = up ? (a < c) : (a > c);
                if (sw) { sk[i] = c; sk[ixj] = a; }
            }
            __syncthreads();
        }
    }
    for (int e = tid; e < tile; e += 1024) keys[base + e] = sk[e];
}

// ---------------- LDS-fused bitonic: merge tail (phases j = tile/2 .. 1 for given k) ----
__global__ __launch_bounds__(1024) void lds_merge_kernel(unsigned long long* __restrict__ keys,
                                                         int P, int k, int tile) {
    __shared__ unsigned long long sk[SORT_TILE];
    const int tid = threadIdx.x;
    const size_t base = (size_t)blockIdx.x * tile;
    const int segBase = (int)(base & (size_t)(P - 1));
    for (int e = tid; e < tile; e += 1024) sk[e] = keys[base + e];
    __syncthreads();
    for (int j = tile >> 1; j >= 1; j >>= 1) {
        for (int p = tid; p < (tile >> 1); p += 1024) {
            int i = ((p & ~(j - 1)) << 1) | (p & (j - 1));
            int ixj = i | j;
            unsigned long long a = sk[i], c = sk[ixj];
            bool up = (((segBase + i) & k) == 0);
            bool sw = up ? (a < c) : (a > c);
            if (sw) { sk[i] = c; sk[ixj] = a; }
        }
        __syncthreads();
    }
    for (int e = tid; e < tile; e += 1024) keys[base + e] = sk[e];
}

// ---------------- global compare-exchange stage (cross-tile j) ----------------
__global__ void bitonic_kernel(unsigned long long* __restrict__ keys,
                               int P, int j, int k, int total) {
    int t = blockIdx.x * blockDim.x + threadIdx.x;
    if (t >= total) return;
    int seg = t / P, i = t - seg * P;
    int ixj = i ^ j;
    if (ixj <= i) return;
    unsigned long long* base = keys + (size_t)seg * P;
    unsigned long long a = base[i], bb = base[ixj];
    bool up = (i & k) == 0;                 // descending region
    bool sw = up ? (a < bb) : (a > bb);
    if (sw) { base[i] = bb; base[ixj] = a; }
}

// ---------------- gather top-1000 per level -> per-image 3000 pool ----------------
__global__ void gather1_kernel(const unsigned long long* __restrict__ k0,
                               const unsigned long long* __restrict__ k1,
                               const unsigned long long* __restrict__ k2,
                               const float* __restrict__ sc, const int* __restrict__ cl,
                               const float* __restrict__ bx,
                               float* __restrict__ gs, float* __restrict__ gc,
                               float* __restrict__ gb) {
    int t = blockIdx.x * blockDim.x + threadIdx.x;
    if (t >= BATCH * NSEL) return;
    int b = t / NSEL, r = t - b * NSEL;
    int lvl = r / TOP_N, rr = r - lvl * TOP_N;
    unsigned long long key; int off;
    if (lvl == 0)      { key = k0[(size_t)b * P0 + rr]; off = 0; }
    else if (lvl == 1) { key = k1[(size_t)b * P1 + rr]; off = N0; }
    else               { key = k2[(size_t)b * P2 + rr]; off = N0 + N1; }
    unsigned idx = 0xFFFFFFFFu - (unsigned)(key & 0xFFFFFFFFull);
    size_t gi = (size_t)b * NTOT + off + idx;
    gs[t] = sc[gi];
    gc[t] = (float)cl[gi];
    ((float4*)gb)[t] = ((const float4*)bx)[gi];
}

// ---------------- global (per-image) sort-key init over 3000 ----------------
__global__ void key2_init_kernel(const float* __restrict__ gs,
                                 unsigned long long* __restrict__ keys) {
    int t = blockIdx.x * blockDim.x + threadIdx.x;
    if (t >= BATCH * PG) return;
    int b = t / PG, p = t - b * PG;
    unsigned long long k = 0ull;
    if (p < NSEL) k = make_key(gs[(size_t)b * NSEL + p], (unsigned)p);
    keys[t] = k;
}

// ---------------- materialize sorted arrays ----------------
__global__ void gather2_kernel(const unsigned long long* __restrict__ keys,
                               const float* __restrict__ gs, const float* __restrict__ gc,
                               const float* __restrict__ gb,
                               float* __restrict__ ss, float* __restrict__ cc,
                               float* __restrict__ bb) {
    int t = blockIdx.x * blockDim.x + threadIdx.x;
    if (t >= BATCH * NSEL) return;
    int b = t / NSEL, r = t - b * NSEL;
    unsigned long long key = keys[(size_t)b * PG + r];
    unsigned idx = 0xFFFFFFFFu - (unsigned)(key & 0xFFFFFFFFull);
    size_t gi = (size_t)b * NSEL + idx;
    ss[t] = gs[gi];
    cc[t] = gc[gi];
    ((float4*)bb)[t] = ((const float4*)gb)[gi];
}

// ---------------- NMS + output compaction: one block per image ----------------
__global__ __launch_bounds__(512) void nms_kernel(const float* __restrict__ s_srt,
                                                  const float* __restrict__ c_srt,
                                                  const float* __restrict__ b_srt,
                                                  float* __restrict__ out) {
    __shared__ unsigned lo[NSEL];      // x1 | y1<<16  (coords in [0,608])
    __shared__ unsigned hi[NSEL];      // x2 | y2<<16
    __shared__ float s_sh[NSEL];
    __shared__ unsigned char keep[NSEL];
    const int b = blockIdx.x, tid = threadIdx.x, bs = blockDim.x;
    const float* sB = s_srt + (size_t)b * NSEL;

#if USE_ASYNC_LDS
    // CDNA5 async global->LDS copy engine; tracked by ASYNCcnt.
    for (int t = tid; t < NSEL; t += bs) {
        __builtin_amdgcn_global_load_async_to_lds_b32(
            (as1_i32p)(sB + t),
            (as3_i32p)(&s_sh[t]),
            0, 0);
    }
#if __has_builtin(__builtin_amdgcn_s_wait_asynccnt)
    __builtin_amdgcn_s_wait_asynccnt(0);
#else
    asm volatile("s_wait_asynccnt 0" ::: "memory");
#endif
#else
    for (int t = tid; t < NSEL; t += bs) s_sh[t] = sB[t];
#endif
    for (int t = tid; t < NSEL; t += bs) {
        float4 bv = ((const float4*)b_srt)[(size_t)b * NSEL + t];
        int x1 = (int)bv.x, y1 = (int)bv.y, x2 = (int)bv.z, y2 = (int)bv.w;
        lo[t] = (unsigned)(x1 & 0xFFFF) | ((unsigned)(y1 & 0xFFFF) << 16);
        hi[t] = (unsigned)(x2 & 0xFFFF) | ((unsigned)(y2 & 0xFFFF) << 16);
    }
    __syncthreads();
    for (int t = tid; t < NSEL; t += bs)
        keep[t] = (s_sh[t] > MIN_SCOREF) ? 1 : 0;
    __syncthreads();

    for (int i = 0; i < NSEL - 1; ++i) {
        if (keep[i]) {                      // uniform LDS read -> uniform branch
            unsigned li = lo[i], hii = hi[i];
            float x1i = (float)(li & 0xFFFF), y1i = (float)(li >> 16);
            float x2i = (float)(hii & 0xFFFF), y2i = (float)(hii >> 16);
            float ai = (x2i - x1i) * (y2i - y1i);
            for (int j = i + 1 + tid; j < NSEL; j += bs) {
                if (!keep[j]) continue;
                unsigned lj = lo[j], hj = hi[j];
                float x1j = (float)(lj & 0xFFFF), y1j = (float)(lj >> 16);
                float x2j = (float)(hj & 0xFFFF), y2j = (float)(hj >> 16);
                float xx1 = fmaxf(x1i, x1j), yy1 = fmaxf(y1i, y1j);
                float xx2 = fminf(x2i, x2j), yy2 = fminf(y2i, y2j);
                float inter = fmaxf(xx2 - xx1, 0.0f) * fmaxf(yy2 - yy1, 0.0f);
                float aj = (x2j - x1j) * (y2j - y1j);
                float iou = inter / (ai + aj - inter);   // 0/0 -> NaN -> no suppress
                if (iou > NMS_THRF) keep[j] = 0;         // unique owner per j
            }
            __syncthreads();
        }
    }

    float* out_s = out;                          // [8][100]
    float* out_c = out + BATCH * MAX_DET;        // [8][100]
    float* out_b = out + 2 * BATCH * MAX_DET;    // [8][100][4]
    for (int t = tid; t < MAX_DET; t += bs) {
        out_s[b * MAX_DET + t] = -1.0f;
        out_c[b * MAX_DET + t] = -1.0f;
    }
    for (int t = tid; t < MAX_DET * 4; t += bs)
        out_b[b * MAX_DET * 4 + t] = -1.0f;
    __syncthreads();
    if (tid == 0) {
        int cnt = 0;
        for (int t = 0; t < NSEL && cnt < MAX_DET; ++t) {
            if (keep[t]) {
                out_s[b * MAX_DET + cnt] = s_sh[t];
                out_c[b * MAX_DET + cnt] = c_srt[(size_t)b * NSEL + t];
                unsigned lt = lo[t], ht = hi[t];
                float* ob = out_b + ((size_t)b * MAX_DET + cnt) * 4;
                ob[0] = (float)(lt & 0xFFFF); ob[1] = (float)(lt >> 16);
                ob[2] = (float)(ht & 0xFFFF); ob[3] = (float)(ht >> 16);
                ++cnt;
            }
        }
    }
}

// Hybrid bitonic: LDS-fused local phases + global cross-tile exchanges.
static inline void run_sort(unsigned long long* keys, int P, hipStream_t stream) {
    const int tile = (P < SORT_TILE) ? P : SORT_TILE;
    const int total = BATCH * P;
    const int tiles = total / tile;
    lds_sort_kernel<<<tiles, 1024, 0, stream>>>(keys, P, tile);   // k = 2 .. tile
    for (int k = tile * 2; k <= P; k <<= 1) {
        for (int j = k >> 1; j >= tile; j >>= 1)                  // cross-tile exchanges
            bitonic_kernel<<<(total + 255) / 256, 256, 0, stream>>>(keys, P, j, k, total);
        lds_merge_kernel<<<tiles, 1024, 0, stream>>>(keys, P, k, tile); // j = tile/2 .. 1
    }
}

extern "C" void kernel_launch(void* const* d_in, const int* in_sizes, int n_in,
                              void* d_out, int out_size, void* d_ws, size_t ws_size,
                              hipStream_t stream) {
    (void)in_sizes; (void)n_in; (void)out_size; (void)ws_size;
    // Input order: obj0, reg0, cls0, anchors0, obj1, reg1, cls1, anchors1, obj2, reg2, cls2, anchors2
    const float* obj[3] = { (const float*)d_in[0], (const float*)d_in[4], (const float*)d_in[8]  };
    const float* reg[3] = { (const float*)d_in[1], (const float*)d_in[5], (const float*)d_in[9]  };
    const float* cls[3] = { (const float*)d_in[2], (const float*)d_in[6], (const float*)d_in[10] };
    const float* anc[3] = { (const float*)d_in[3], (const float*)d_in[7], (const float*)d_in[11] };

    char* w = (char*)d_ws; size_t off = 0;
    auto wsalloc = [&](size_t bytes) -> void* {
        void* p = w + off; off = (off + bytes + 255) & ~(size_t)255; return p;
    };
    float*              sc = (float*)             wsalloc((size_t)BATCH * NTOT * 4);
    int*                cl = (int*)               wsalloc((size_t)BATCH * NTOT * 4);
    float*              bx = (float*)             wsalloc((size_t)BATCH * NTOT * 16);
    unsigned long long* k0 = (unsigned long long*)wsalloc((size_t)BATCH * P0 * 8);
    unsigned long long* k1 = (unsigned long long*)wsalloc((size_t)BATCH * P1 * 8);
    unsigned long long* k2 = (unsigned long long*)wsalloc((size_t)BATCH * P2 * 8);
    float*              gs = (float*)             wsalloc((size_t)BATCH * NSEL * 4);
    float*              gc = (float*)             wsalloc((size_t)BATCH * NSEL * 4);
    float*              gb = (float*)             wsalloc((size_t)BATCH * NSEL * 16);
    unsigned long long* k3 = (unsigned long long*)wsalloc((size_t)BATCH * PG * 8);
    float*              ss = (float*)             wsalloc((size_t)BATCH * NSEL * 4);
    float*              cc = (float*)             wsalloc((size_t)BATCH * NSEL * 4);
    float*              bb = (float*)             wsalloc((size_t)BATCH * NSEL * 16);

    const int Ns[3]   = { N0, N1, N2 };
    const int Offs[3] = { 0, N0, N0 + N1 };
    for (int l = 0; l < 3; ++l) {
        int total = BATCH * Ns[l];
        decode_kernel<<<(total + 255) / 256, 256, 0, stream>>>(
            obj[l], reg[l], cls[l], anc[l], sc, cl, bx, Ns[l], Offs[l]);
    }
    key_init_kernel<<<(BATCH * P0 + 255) / 256, 256, 0, stream>>>(sc, k0, P0, N0, Offs[0]);
    key_init_kernel<<<(BATCH * P1 + 255) / 256, 256, 0, stream>>>(sc, k1, P1, N1, Offs[1]);
    key_init_kernel<<<(BATCH * P2 + 255) / 256, 256, 0, stream>>>(sc, k2, P2, N2, Offs[2]);
    run_sort(k0, P0, stream);   // 1 launch
    run_sort(k1, P1, stream);   // 3 launches
    run_sort(k2, P2, stream);   // 10 launches

    gather1_kernel<<<(BATCH * NSEL + 255) / 256, 256, 0, stream>>>(
        k0, k1, k2, sc, cl, bx, gs, gc, gb);

    key2_init_kernel<<<(BATCH * PG + 255) / 256, 256, 0, stream>>>(gs, k3);
    run_sort(k3, PG, stream);   // 1 launch

    gather2_kernel<<<(BATCH * NSEL + 255) / 256, 256, 0, stream>>>(
        k3, gs, gc, gb, ss, cc, bb);

    nms_kernel<<<BATCH, 512, 0, stream>>>(ss, cc, bb, (float*)d_out);
}